// BahdanauAttention_79405355368720
// MI455X (gfx1250) — compile-verified
//
#include <hip/hip_runtime.h>
#include <hip/hip_bf16.h>

// Reference collapses: softmax over a singleton axis == 1.0 exactly, so
//   attention_weights[b,t,0] = 1.0f
//   coverage_vector[b,t,0]   = (float)t
//   context_vector[b,d]      = sum_t values[b,t,d]
// All GEMMs (values@W2, query@W1, tanh, @V, @W3) are dead code.
// Live work = stream 128 MB of `values` once -> bandwidth bound (~5.5us @ 23.3 TB/s).

typedef __attribute__((ext_vector_type(2))) float v2f;
typedef __attribute__((ext_vector_type(8))) float v8f;

#define BB 32
#define TT 1024
#define DD 1024

// ---- aux outputs: attention weights (all ones) and coverage (t) -------------
__global__ void bahdanau_aux_fill(float* __restrict__ aw, float* __restrict__ cov) {
    int i = blockIdx.x * blockDim.x + threadIdx.x;   // 0 .. B*T-1
    if (i < BB * TT) {
        aw[i]  = 1.0f;
        cov[i] = (float)(i & (TT - 1));              // t = i % T (T power of two)
    }
}

// ---- context_vector[b, d] = sum_t values[b, t, d] via fp32 WMMA -------------
// One wave per (b, 16-wide d-tile). A = ones(16x4), B = values tile (4 t x 16 d),
// D = A*B + C accumulates exact fp32 column sums over T in K=4 chunks.
// B-matrix VGPR layout (4x16 f32, row striped across lanes per VGPR):
//   VGPR0: lanes 0-15 -> row K=0, lanes 16-31 -> row K=2
//   VGPR1: lanes 0-15 -> row K=1, lanes 16-31 -> row K=3
// Since A is all-ones, any t-to-K-slot assignment is mathematically identical;
// only the column mapping N = lane&15 must match the d index.
__global__ __launch_bounds__(256) void bahdanau_context_sum_wmma(
        const float* __restrict__ values, float* __restrict__ ctx) {
    const int lane  = threadIdx.x & 31;
    const int wave  = threadIdx.x >> 5;
    const int tile  = blockIdx.x * 8 + wave;     // 0 .. B*(D/16)-1 = 2047
    const int b     = tile >> 6;                 // tile / (D/16)
    const int dbase = (tile & 63) << 4;          // 16-wide d tile
    const int h     = lane >> 4;                 // half-wave select
    const int c     = lane & 15;                 // d column within tile

    const float* base = values + (size_t)b * TT * DD + dbase + c;

    v2f a;                                       // A = ones(16x4): 2 VGPRs/lane
    a[0] = 1.0f; a[1] = 1.0f;
    v8f acc = {};                                // C/D accumulator: 8 VGPRs

#pragma unroll 4
    for (int t0 = 0; t0 < TT; t0 += 4) {
        v2f bt;
        bt[0] = base[(size_t)(t0 + 2 * h)     * DD];   // VGPR0: rows {0,2}
        bt[1] = base[(size_t)(t0 + 2 * h + 1) * DD];   // VGPR1: rows {1,3}
        acc = __builtin_amdgcn_wmma_f32_16x16x4_f32(
                /*neg_a=*/false, a, /*neg_b=*/false, bt,
                /*c_mod=*/(short)0, acc, /*reuse_a=*/false, /*reuse_b=*/false);
    }

    // D row 0 lives in acc[0] of lanes 0-15 (M=0, N=lane); each column holds
    // sum_t values[b, t, dbase+N].
    if (lane < 16) {
        ctx[(size_t)b * DD + dbase + c] = acc[0];
    }
}

extern "C" void kernel_launch(void* const* d_in, const int* in_sizes, int n_in,
                              void* d_out, int out_size, void* d_ws, size_t ws_size,
                              hipStream_t stream) {
    // inputs: 0=query 1=values 2=W1 3=b1 4=W2 5=b2 6=W3 7=b3 8=V 9=bV
    const float* values = (const float*)d_in[1];

    float* out = (float*)d_out;
    float* ctx = out;                 // [B, D]    = 32768 floats
    float* aw  = out + BB * DD;       // [B, T, 1] = 32768 floats
    float* cov = aw + BB * TT;        // [B, T, 1] = 32768 floats

    hipLaunchKernelGGL(bahdanau_aux_fill,
                       dim3((BB * TT + 255) / 256), dim3(256), 0, stream, aw, cov);

    // B*(D/16) = 2048 waves, 8 waves (256 threads) per block -> 256 blocks
    hipLaunchKernelGGL(bahdanau_context_sum_wmma,
                       dim3(256), dim3(256), 0, stream, values, ctx);
}